// DGALoss_60601988547134
// MI455X (gfx1250) — compile-verified
//
#include <hip/hip_runtime.h>
#include <hip/hip_bf16.h>
#include <math.h>

// Problem constants (match reference setup_inputs: B=32, N=65536)
#define NB     32
#define NN     65536
#define DT_C   0.005f
#define INV_HUBER 200.0f   // 1/0.005
#define N0_C   5

typedef __attribute__((ext_vector_type(2))) float v2f;
typedef __attribute__((ext_vector_type(8))) float v8f;

// ---------------------------------------------------------------------------
// Wave32 full-sum via V_WMMA_F32_16X16X4_F32 (ones-matmul reduction).
// A layout (ISA 7.12.2, 32-bit A 16x4): lane L<16 holds A[L][0],A[L][1];
// lane L+16 holds A[L][2],A[L][3].  Put v in VGPR0, 0 in VGPR1 =>
// row L sum = v_L + v_{L+16} (B = all-ones is layout-independent).
// D layout: VGPR m, lanes<16 -> D[m][n], lanes>=16 -> D[m+8][n-16].
// Sum of 8 D regs per lane + shfl_xor(16) == exact fp32 wave sum, all lanes.
// ---------------------------------------------------------------------------
#if defined(__has_builtin)
#if __has_builtin(__builtin_amdgcn_wmma_f32_16x16x4_f32)
#define HAVE_WMMA_X4 1
#endif
#endif

__device__ __forceinline__ float wave_sum32(float v) {
#ifdef HAVE_WMMA_X4
  v2f a; a[0] = v;    a[1] = 0.0f;
  v2f b; b[0] = 1.0f; b[1] = 1.0f;
  v8f c = {};
  c = __builtin_amdgcn_wmma_f32_16x16x4_f32(false, a, false, b,
                                            (short)0, c, false, false);
  float s = c[0] + c[1] + c[2] + c[3] + c[4] + c[5] + c[6] + c[7];
  s += __shfl_xor(s, 16, 32);
  return s;
#else
#pragma unroll
  for (int d = 16; d >= 1; d >>= 1) v += __shfl_xor(v, d, 32);
  return v;
#endif
}

// ---------------------------------------------------------------------------
// 3x3 SO(3) helpers
// ---------------------------------------------------------------------------
struct M3 { float m[9]; };

__device__ __forceinline__ M3 so3exp(float x, float y, float z) {
  float t2 = x * x + y * y + z * z;
  float A, B;
  if (t2 < 1e-12f) {
    A = 1.0f - t2 * (1.0f / 6.0f);
    B = 0.5f - t2 * (1.0f / 24.0f);
  } else {
    float t = sqrtf(t2 + 1e-24f);
    A = sinf(t) / t;
    B = (1.0f - cosf(t)) / (t2 + 1e-24f);
  }
  M3 R;
  R.m[0] = 1.0f + B * (x * x - t2);
  R.m[1] = -A * z + B * x * y;
  R.m[2] =  A * y + B * x * z;
  R.m[3] =  A * z + B * y * x;
  R.m[4] = 1.0f + B * (y * y - t2);
  R.m[5] = -A * x + B * y * z;
  R.m[6] = -A * y + B * z * x;
  R.m[7] =  A * x + B * z * y;
  R.m[8] = 1.0f + B * (z * z - t2);
  return R;
}

__device__ __forceinline__ M3 mul3(const M3& a, const M3& b) {
  M3 c;
#pragma unroll
  for (int i = 0; i < 3; i++)
#pragma unroll
    for (int j = 0; j < 3; j++)
      c.m[i * 3 + j] = a.m[i * 3 + 0] * b.m[0 * 3 + j] +
                       a.m[i * 3 + 1] * b.m[1 * 3 + j] +
                       a.m[i * 3 + 2] * b.m[2 * 3 + j];
  return c;
}

// a^T * b
__device__ __forceinline__ M3 mulTN(const M3& a, const M3& b) {
  M3 c;
#pragma unroll
  for (int i = 0; i < 3; i++)
#pragma unroll
    for (int j = 0; j < 3; j++)
      c.m[i * 3 + j] = a.m[0 * 3 + i] * b.m[0 * 3 + j] +
                       a.m[1 * 3 + i] * b.m[1 * 3 + j] +
                       a.m[2 * 3 + i] * b.m[2 * 3 + j];
  return c;
}

__device__ __forceinline__ float huber1(float r) {
  float x = r * INV_HUBER;
  float ax = fabsf(x);
  return (ax < 1.0f) ? 0.5f * x * x : ax - 0.5f;
}

// SmoothL1(so3_log(R)) summed over the 3 components
__device__ __forceinline__ float huber_of_log(const M3& R) {
  float tr = R.m[0] + R.m[4] + R.m[8];
  float cs = (tr - 1.0f) * 0.5f;
  cs = fminf(fmaxf(cs, -1.0f + 1e-7f), 1.0f - 1e-7f);
  float th = acosf(cs);
  float coef = (th < 1e-6f) ? (0.5f + th * th * (1.0f / 12.0f))
                            : (th / (2.0f * sinf(th)));
  float vx = coef * (R.m[7] - R.m[5]);
  float vy = coef * (R.m[2] - R.m[6]);
  float vz = coef * (R.m[3] - R.m[1]);
  return huber1(vx) + huber1(vy) + huber1(vz);
}

// Left-to-right product of 16 consecutive exp(DT*w) (== the pairwise tree)
__device__ __forceinline__ M3 chain16(const float* __restrict__ p) {
  M3 P = so3exp(p[0] * DT_C, p[1] * DT_C, p[2] * DT_C);
#pragma unroll 4
  for (int j = 1; j < 16; j++) {
    M3 R = so3exp(p[3 * j + 0] * DT_C, p[3 * j + 1] * DT_C, p[3 * j + 2] * DT_C);
    P = mul3(P, R);
  }
  return P;
}

// ---------------------------------------------------------------------------
// Gyro kernel: one thread per 32-sample group (65536 threads total)
// ---------------------------------------------------------------------------
__global__ __launch_bounds__(256) void gyro_kernel(const float* __restrict__ w_hat,
                                                   const float* __restrict__ xs,
                                                   float* __restrict__ part16,
                                                   float* __restrict__ part32) {
  int g = blockIdx.x * 256 + threadIdx.x;            // 0 .. 65535
  const float* wp = w_hat + (size_t)g * 96;          // 32 samples * 3
  const float* xp = xs + (size_t)g * 96;

  __builtin_prefetch(xp, 0, 3);                      // global_prefetch_b8
  __builtin_prefetch(wp + 48, 0, 3);

  M3 P16a = chain16(wp);
  M3 P16b = chain16(wp + 48);
  M3 P32  = mul3(P16a, P16b);

  M3 Xa  = so3exp(xp[0],  xp[1],  xp[2]);            // xs flat element 32g
  M3 Xb  = so3exp(xp[48], xp[49], xp[50]);           // xs flat element 32g+16
  M3 X32 = mul3(Xa, Xb);

  float la = huber_of_log(mulTN(P16a, Xa));
  float lb = huber_of_log(mulTN(P16b, Xb));
  float lc = huber_of_log(mulTN(P32,  X32));

  int m16a = (2 * g) & 4095;                         // row index within batch (16-level)
  int m32  = g & 2047;                               // row index within batch (32-level)
  float s16 = 0.0f, s32 = 0.0f;
  if (m16a     >= N0_C) s16 += la;
  if (m16a + 1 >= N0_C) s16 += lb;
  if (m32      >= N0_C) s32 += lc;

  float t16 = wave_sum32(s16);
  float t32 = wave_sum32(s32);

  __shared__ float l16[8], l32[8];
  int lane = threadIdx.x & 31, wid = threadIdx.x >> 5;
  if (lane == 0) { l16[wid] = t16; l32[wid] = t32; }
  __syncthreads();
  if (threadIdx.x == 0) {
    float a = 0.0f, b = 0.0f;
#pragma unroll
    for (int i = 0; i < 8; i++) { a += l16[i]; b += l32[i]; }
    part16[blockIdx.x] = a;
    part32[blockIdx.x] = b;
  }
}

// ---------------------------------------------------------------------------
// Velocity kernel: one 1024-thread block per batch.
// Chained block scans: S = cumsum(a), vs = DT*(2S - a - a0), T = cumsum(vs).
// vs[i] = T[i]-T[i-1]; window mean = (T[i]-T[i-16])/16. T tiles live in LDS.
// ---------------------------------------------------------------------------
struct F3 { float x, y, z; };

__device__ __forceinline__ F3 wave_scan3(F3 v, int lane) {
#pragma unroll
  for (int d = 1; d < 32; d <<= 1) {
    float ax = __shfl_up(v.x, d, 32);
    float ay = __shfl_up(v.y, d, 32);
    float az = __shfl_up(v.z, d, 32);
    if (lane >= d) { v.x += ax; v.y += ay; v.z += az; }
  }
  return v;
}

__device__ __forceinline__ F3 block_scan3(F3 v, F3* wsum) {
  int lane = threadIdx.x & 31, wid = threadIdx.x >> 5;
  v = wave_scan3(v, lane);
  if (lane == 31) wsum[wid] = v;
  __syncthreads();
  if (wid == 0) {
    F3 w = wsum[lane];
    w = wave_scan3(w, lane);
    wsum[lane] = w;
  }
  __syncthreads();
  if (wid > 0) {
    F3 o = wsum[wid - 1];
    v.x += o.x; v.y += o.y; v.z += o.z;
  }
  __syncthreads();   // make wsum reusable
  return v;
}

__global__ __launch_bounds__(1024) void vel_kernel(const float* __restrict__ a_hat,
                                                   const float* __restrict__ gt,
                                                   float* __restrict__ partsq) {
  __shared__ F3 wsum[32];
  __shared__ F3 tileT[1024];
  __shared__ F3 tail[16];
  __shared__ F3 sLast;

  int b = blockIdx.x;
  int tid = threadIdx.x;
  const float* ab = a_hat + (size_t)b * NN * 3;
  const float* gb = gt    + (size_t)b * NN * 3;
  F3 a0; a0.x = ab[0]; a0.y = ab[1]; a0.z = ab[2];

  if (tid < 16) { tail[tid].x = 0.f; tail[tid].y = 0.f; tail[tid].z = 0.f; }
  __syncthreads();

  F3 Sc = {0.f, 0.f, 0.f}, Tc = {0.f, 0.f, 0.f};
  float sq = 0.0f;

  for (int tile = 0; tile < NN / 1024; ++tile) {
    int idx = tile * 1024 + tid;
    F3 a; a.x = ab[idx * 3 + 0]; a.y = ab[idx * 3 + 1]; a.z = ab[idx * 3 + 2];

    F3 s = block_scan3(a, wsum);
    F3 S; S.x = Sc.x + s.x; S.y = Sc.y + s.y; S.z = Sc.z + s.z;
    F3 vs;
    vs.x = DT_C * (2.0f * S.x - a.x - a0.x);
    vs.y = DT_C * (2.0f * S.y - a.y - a0.y);
    vs.z = DT_C * (2.0f * S.z - a.z - a0.z);
    if (tid == 1023) sLast = S;

    F3 t = block_scan3(vs, wsum);
    F3 T; T.x = Tc.x + t.x; T.y = Tc.y + t.y; T.z = Tc.z + t.z;
    tileT[tid] = T;
    __syncthreads();

    F3 Tm1  = (tid >= 1)  ? tileT[tid - 1]  : tail[15];
    F3 Tm16 = (tid >= 16) ? tileT[tid - 16] : tail[tid];
    F3 vsn;
    vsn.x = (T.x - Tm1.x) - (T.x - Tm16.x) * (1.0f / 16.0f);
    vsn.y = (T.y - Tm1.y) - (T.y - Tm16.y) * (1.0f / 16.0f);
    vsn.z = (T.z - Tm1.z) - (T.z - Tm16.z) * (1.0f / 16.0f);
    float dx = gb[idx * 3 + 0] - vsn.x;
    float dy = gb[idx * 3 + 1] - vsn.y;
    float dz = gb[idx * 3 + 2] - vsn.z;
    sq += dx * dx + dy * dy + dz * dz;

    F3 nS = sLast;
    F3 nT = tileT[1023];
    __syncthreads();
    if (tid < 16) tail[tid] = tileT[1008 + tid];
    __syncthreads();
    Sc = nS; Tc = nT;
  }

  float tw = wave_sum32(sq);
  __shared__ float wpart[32];
  int lane = tid & 31, wid = tid >> 5;
  if (lane == 0) wpart[wid] = tw;
  __syncthreads();
  if (tid == 0) {
    float acc = 0.0f;
#pragma unroll
    for (int i = 0; i < 32; i++) acc += wpart[i];
    partsq[b] = acc;
  }
}

// ---------------------------------------------------------------------------
// Final deterministic combine -> scalar
// ---------------------------------------------------------------------------
__global__ __launch_bounds__(256) void final_kernel(const float* __restrict__ part16,
                                                    const float* __restrict__ part32,
                                                    const float* __restrict__ partsq,
                                                    float* __restrict__ out) {
  int tid = threadIdx.x;
  float a16 = part16[tid];                     // 256 gyro blocks
  float a32 = part32[tid];
  float asq = (tid < NB) ? partsq[tid] : 0.0f; // 32 vel blocks

  float t16 = wave_sum32(a16);
  float t32 = wave_sum32(a32);
  float tsq = wave_sum32(asq);

  __shared__ float l1[8], l2[8], l3[8];
  int lane = tid & 31, wid = tid >> 5;
  if (lane == 0) { l1[wid] = t16; l2[wid] = t32; l3[wid] = tsq; }
  __syncthreads();
  if (tid == 0) {
    float S16 = 0.f, S32 = 0.f, Ssq = 0.f;
#pragma unroll
    for (int i = 0; i < 8; i++) { S16 += l1[i]; S32 += l2[i]; Ssq += l3[i]; }
    // f_huber = 1e6 * mean * 0.005^2 = 25 * mean ; 32-level additionally / 2
    float loss16 = 25.0f * S16 / (float)(NB * 4091 * 3);
    float loss32 = 12.5f * S32 / (float)(NB * 2043 * 3);
    float lacc   = Ssq / ((float)NB * (float)NN * 3.0f);
    out[0] = loss16 + loss32 + lacc;
  }
}

// ---------------------------------------------------------------------------
extern "C" void kernel_launch(void* const* d_in, const int* in_sizes, int n_in,
                              void* d_out, int out_size, void* d_ws, size_t ws_size,
                              hipStream_t stream) {
  const float* w_hat = (const float*)d_in[0];
  const float* a_hat = (const float*)d_in[1];
  const float* xs    = (const float*)d_in[2];
  // d_in[3] (dv) is unused by the reference
  const float* vs_gt = (const float*)d_in[4];

  float* ws = (float*)d_ws;
  float* part16 = ws;           // 256 floats
  float* part32 = ws + 256;     // 256 floats
  float* partsq = ws + 512;     // 32 floats

  gyro_kernel<<<256, 256, 0, stream>>>(w_hat, xs, part16, part32);
  vel_kernel<<<NB, 1024, 0, stream>>>(a_hat, vs_gt, partsq);
  final_kernel<<<1, 256, 0, stream>>>(part16, part32, partsq, (float*)d_out);
}